// SnakeFastLoss_7344394076748
// MI455X (gfx1250) — compile-verified
//
#include <hip/hip_runtime.h>
#include <math.h>

// ---------------------------------------------------------------------------
// SnakeFastLoss for MI455X (gfx1250, wave32).
// Pipeline (all on `stream`, data stays resident in L2 — total footprint <5MB):
//   1) grad_conv:  direct 13x13 Gaussian-derivative conv on pred and |pred|
//   2) snake:      50 sequential steps, 1 block per batch, LDS neighbor sync
//   3) render+loss fused: per-pixel min over 256 nodes of sqrt(d2)-w via
//      V_WMMA_F32_16X16X4_F32 (d2 as a rank-4 bilinear form), then MSE reduce.
// Workspace layout (floats): gimg[B*2*H*W] | gimgW[B*2*H*W] | pos[B*N*2] | w[B*N]
// ---------------------------------------------------------------------------

static constexpr int kB = 4;
static constexpr int kH = 256;
static constexpr int kW = 256;
static constexpr int kN = 256;   // snake control points
static constexpr float kStep   = 0.1f;
static constexpr float kAlpha  = 0.01f;
static constexpr float kBeta   = 0.005f;
static constexpr float kDmax   = 15.0f;
static constexpr float kExtFac = 10.0f;

typedef float v2f __attribute__((ext_vector_type(2)));
typedef float v8f __attribute__((ext_vector_type(8)));

// ---------------------------------------------------------------------------
__global__ void zero_out_kernel(float* out) { out[0] = 0.0f; }

// ---------------------------------------------------------------------------
// Kernel 1: gradient images. One thread per (b, y, x) pixel; computes both
// channels of gimg (from pred) and gimgW (from |pred|) in one pass.
// fy[i][j] = dg[i]*g[j], fx[i][j] = g[i]*dg[j]; SAME zero padding; corr (no flip).
__global__ void grad_conv_kernel(const float* __restrict__ pred,
                                 float* __restrict__ gimg,
                                 float* __restrict__ gimgW) {
    int idx = blockIdx.x * blockDim.x + threadIdx.x;
    if (idx >= kB * kH * kW) return;
    int b   = idx >> 16;
    int pix = idx & 0xFFFF;
    int y   = pix >> 8;
    int x   = pix & 255;

    // Rebuild the 13-tap normalized Gaussian + derivative (stdev=2).
    float g[13], dg[13];
    float s = 0.0f;
#pragma unroll
    for (int k = 0; k < 13; ++k) {
        float t = (float)(k - 6);
        g[k] = __expf(-t * t * 0.125f);   // exp(-t^2 / (2*stdev^2)), stdev=2
        s += g[k];
    }
    float inv = 1.0f / s;
#pragma unroll
    for (int k = 0; k < 13; ++k) {
        g[k] *= inv;
        dg[k] = -((float)(k - 6)) * 0.25f * g[k];   // -t/stdev^2 * g
    }

    const float* in = pred + b * kH * kW;
    float afy = 0.f, afx = 0.f, wfy = 0.f, wfx = 0.f;
    for (int i = 0; i < 13; ++i) {
        int yy = y + i - 6;
        if (yy < 0 || yy >= kH) continue;
        const float* row = in + yy * kW;
        float gi = g[i], dgi = dg[i];
#pragma unroll
        for (int j = 0; j < 13; ++j) {
            int xx = x + j - 6;
            if (xx < 0 || xx >= kW) continue;
            float v  = row[xx];
            float av = fabsf(v);
            float wy = dgi * g[j];   // fy weight
            float wx = gi * dg[j];   // fx weight
            afy += v * wy;  afx += v * wx;
            wfy += av * wy; wfx += av * wx;
        }
    }
    float* go = gimg  + (b * 2) * kH * kW;
    float* gw = gimgW + (b * 2) * kH * kW;
    go[pix]            = afy * kExtFac;
    go[kH * kW + pix]  = afx * kExtFac;
    gw[pix]            = wfy * kExtFac;
    gw[kH * kW + pix]  = wfx * kExtFac;
}

// ---------------------------------------------------------------------------
// Kernel 2: snake optimization. 1 block per batch, thread i owns point i.
// 50 serial steps; neighbor values exchanged through LDS (4 barriers/step).
__global__ void snake_kernel(const float* __restrict__ pos0,
                             const float* __restrict__ w0,
                             const float* __restrict__ gimg,
                             const float* __restrict__ gimgW,
                             float* __restrict__ posOut,
                             float* __restrict__ wOut) {
    __shared__ float sy[kN], sx[kN], sw[kN], t2y[kN], t2x[kN];
    int b = blockIdx.x;
    int i = threadIdx.x;
    const float* gi0 = gimg  + (b * 2) * kH * kW;
    const float* gi1 = gi0 + kH * kW;
    const float* gw0 = gimgW + (b * 2) * kH * kW;
    const float* gw1 = gw0 + kH * kW;

    float py = pos0[(b * kN + i) * 2 + 0];
    float px = pos0[(b * kN + i) * 2 + 1];
    float wd = w0[b * kN + i];
    sy[i] = py; sx[i] = px; sw[i] = wd;
    __syncthreads();

    int im1 = (i > 0) ? i - 1 : 0;
    int ip1 = (i < kN - 1) ? i + 1 : kN - 1;
    const float hiY = 254.999f;   // H - 1.001
    const float hiX = 254.999f;   // W - 1.001

    for (int step = 0; step < 50; ++step) {
        // Second differences from pre-step state.
        float d2y = sy[im1] - 2.f * py + sy[ip1];
        float d2x = sx[im1] - 2.f * px + sx[ip1];
        float d2w = sw[im1] - 2.f * wd + sw[ip1];

        // f = bilinear(gimg, p) at pre-step p.
        float cy = fminf(fmaxf(py, 0.f), hiY);
        float cx = fminf(fmaxf(px, 0.f), hiX);
        int y0 = (int)floorf(cy), x0 = (int)floorf(cx);
        float ty = cy - (float)y0, tx = cx - (float)x0;
        int o = y0 * kW + x0;
        float w00 = (1.f - ty) * (1.f - tx), w01 = (1.f - ty) * tx;
        float w10 = ty * (1.f - tx),         w11 = ty * tx;
        float f0 = gi0[o] * w00 + gi0[o + 1] * w01 + gi0[o + kW] * w10 + gi0[o + kW + 1] * w11;
        float f1 = gi1[o] * w00 + gi1[o + 1] * w01 + gi1[o + kW] * w10 + gi1[o + kW + 1] * w11;

        __syncthreads();             // everyone done reading sy/sx/sw
        t2y[i] = d2y; t2x[i] = d2x;
        __syncthreads();
        float d4y = t2y[im1] - 2.f * d2y + t2y[ip1];
        float d4x = t2x[im1] - 2.f * d2x + t2x[ip1];

        py = fminf(fmaxf(py + kStep * (kAlpha * d2y - kBeta * d4y + f0), 0.f), 255.f);
        px = fminf(fmaxf(px + kStep * (kAlpha * d2x - kBeta * d4x + f1), 0.f), 255.f);

        // fw = bilinear(gimgW, p_new).sum(channels)
        cy = fminf(fmaxf(py, 0.f), hiY);
        cx = fminf(fmaxf(px, 0.f), hiX);
        y0 = (int)floorf(cy); x0 = (int)floorf(cx);
        ty = cy - (float)y0; tx = cx - (float)x0;
        o  = y0 * kW + x0;
        w00 = (1.f - ty) * (1.f - tx); w01 = (1.f - ty) * tx;
        w10 = ty * (1.f - tx);          w11 = ty * tx;
        float fw = (gw0[o] + gw1[o]) * w00 + (gw0[o + 1] + gw1[o + 1]) * w01 +
                   (gw0[o + kW] + gw1[o + kW]) * w10 + (gw0[o + kW + 1] + gw1[o + kW + 1]) * w11;
        wd = fminf(fmaxf(wd + kStep * (kAlpha * d2w + fw), 0.f), kDmax);

        __syncthreads();
        sy[i] = py; sx[i] = px; sw[i] = wd;
        __syncthreads();
    }

    posOut[(b * kN + i) * 2 + 0] = py;
    posOut[(b * kN + i) * 2 + 1] = px;
    wOut[b * kN + i] = wd;
}

// ---------------------------------------------------------------------------
// Kernel 3: fused render + MSE loss.
// d2(m,n) = (ym^2+xm^2) + (pyn^2+pxn^2) - 2(ym*pyn + xm*pxn)
//        = A(16x4) x B(4x16) with A=[y,x,y^2+x^2,1], B=[-2py,-2px,1,py^2+px^2]
// Each wave: 16 pixels; 16 WMMAs cover all 256 nodes. The B fragments are
// stored PRE-SWIZZLED in LDS (indexed [group*32 + lane]) so the per-wave
// preload is a uniform straight-line ds_load stream — no exec-masked loads.
// The group loop is fully unrolled: 16 independent WMMAs overlap with the
// VALU sqrt/min tails. Native v_sqrt_f32 is safe: d2 in [0, ~1.3e5].
__global__ void render_loss_kernel(const float* __restrict__ pred,
                                   const float* __restrict__ pos,
                                   const float* __restrict__ wnode,
                                   float* __restrict__ out) {
    __shared__ float sbx[512], sby[512];   // swizzled B fragments [g*32 + lane]
    __shared__ float nw[kN];
    __shared__ float blockAcc;

    int b       = blockIdx.x >> 9;     // 512 blocks per batch
    int tileInB = blockIdx.x & 511;    // 128 pixels per block
    int t       = threadIdx.x;
    if (t == 0) blockAcc = 0.0f;

    // Build swizzled B fragments: entry (g, l) holds what lane l of any wave
    // needs for node group g.  lanes 0-15: K=0,1 rows (-2py, -2px);
    // lanes 16-31: K=2,3 rows (1, py^2+px^2).  Select is a register cndmask,
    // not a guarded load.
#pragma unroll
    for (int e = t; e < 512; e += 256) {
        int gl = e & 31;                 // lane slot
        int n  = ((e >> 5) << 4) + (gl & 15);   // node index
        float py = pos[(b * kN + n) * 2 + 0];
        float px = pos[(b * kN + n) * 2 + 1];
        bool lo = (gl < 16);
        sbx[e] = lo ? (-2.f * py) : 1.0f;
        sby[e] = lo ? (-2.f * px) : (py * py + px * px);
    }
    nw[t] = wnode[b * kN + t];
    __syncthreads();

    int lane   = t & 31;
    int waveId = t >> 5;
    int basePix = tileInB * 128 + waveId * 16;
    int m = lane & 15;
    int pix = basePix + m;
    float yf = (float)(pix >> 8);
    float xf = (float)(pix & 255);

    // A-matrix 16x4 f32: lanes 0-15 hold K=0,1; lanes 16-31 hold K=2,3.
    v2f a;
    if (lane < 16) { a.x = yf;                a.y = xf;  }
    else           { a.x = yf * yf + xf * xf; a.y = 1.f; }

    // Uniform straight-line preload of all 16 B fragments + widths.
    float bx[16], by[16], wv[16];
#pragma unroll
    for (int g = 0; g < 16; ++g) {
        bx[g] = sbx[g * 32 + lane];
        by[g] = sby[g * 32 + lane];
        wv[g] = nw[g * 16 + m];
    }

    float mn[8];
#pragma unroll
    for (int r = 0; r < 8; ++r) mn[r] = 3.0e38f;

#pragma unroll
    for (int g = 0; g < 16; ++g) {
        v2f bv;
        bv.x = bx[g];
        bv.y = by[g];
        v8f c = {0.f, 0.f, 0.f, 0.f, 0.f, 0.f, 0.f, 0.f};
        c = __builtin_amdgcn_wmma_f32_16x16x4_f32(
                /*neg_a=*/false, a, /*neg_b=*/false, bv,
                /*c_mod=*/(short)0, c, /*reuse_a=*/false, /*reuse_b=*/false);
        float wn = wv[g];
#pragma unroll
        for (int r = 0; r < 8; ++r) {
            float d2 = fmaxf(c[r], 0.f);                    // guard FP cancellation
            float cand = __builtin_amdgcn_sqrtf(d2) - wn;   // raw v_sqrt_f32
            mn[r] = fminf(mn[r], cand);
        }
    }

    // Min across the 16 lanes of each half (N dimension of the C/D tile).
#pragma unroll
    for (int r = 0; r < 8; ++r) {
        float v = mn[r];
        v = fminf(v, __shfl_xor(v, 1, 32));
        v = fminf(v, __shfl_xor(v, 2, 32));
        v = fminf(v, __shfl_xor(v, 4, 32));
        v = fminf(v, __shfl_xor(v, 8, 32));
        mn[r] = v;
    }

    float lsum = 0.f;
    if (m == 0) {   // lane 0 -> pixels M=0..7, lane 16 -> pixels M=8..15
        int rowOff = (lane >> 4) * 8;
        const float* pb = pred + b * kH * kW;
#pragma unroll
        for (int r = 0; r < 8; ++r) {
            int p2 = basePix + rowOff + r;
            float dm = fminf(fmaxf(mn[r], 0.f), kDmax);
            float diff = pb[p2] - dm;
            lsum += diff * diff;
        }
        atomicAdd(&blockAcc, lsum);
    }
    __syncthreads();
    if (t == 0) atomicAdd(out, blockAcc * (1.0f / (float)(kB * kH * kW)));
}

// ---------------------------------------------------------------------------
extern "C" void kernel_launch(void* const* d_in, const int* in_sizes, int n_in,
                              void* d_out, int out_size, void* d_ws, size_t ws_size,
                              hipStream_t stream) {
    (void)in_sizes; (void)n_in; (void)out_size; (void)ws_size;
    const float* pred = (const float*)d_in[0];   // (B,1,H,W) f32
    const float* pos  = (const float*)d_in[1];   // (B,N,2)   f32
    const float* wid  = (const float*)d_in[2];   // (B,N)     f32
    float* out = (float*)d_out;                  // scalar f32

    float* ws    = (float*)d_ws;
    float* gimg  = ws;                           // B*2*H*W
    float* gimgW = gimg  + kB * 2 * kH * kW;     // B*2*H*W
    float* posF  = gimgW + kB * 2 * kH * kW;     // B*N*2
    float* wF    = posF  + kB * kN * 2;          // B*N

    zero_out_kernel<<<1, 1, 0, stream>>>(out);
    grad_conv_kernel<<<(kB * kH * kW + 255) / 256, 256, 0, stream>>>(pred, gimg, gimgW);
    snake_kernel<<<kB, kN, 0, stream>>>(pos, wid, gimg, gimgW, posF, wF);
    render_loss_kernel<<<kB * 512, 256, 0, stream>>>(pred, posF, wF, out);
}